// HungarianMatcher_56367150793468
// MI455X (gfx1250) — compile-verified
//
#include <hip/hip_runtime.h>
#include <hip/hip_bf16.h>

typedef __attribute__((ext_vector_type(2))) float v2f;
typedef __attribute__((ext_vector_type(8))) float v8f;

#define BS 16
#define NQ 625
#define KP 45
#define NTGT 320
#define M_TOTAL (BS * NQ)              // 10000 pred rows
#define TILES_M (M_TOTAL / 16)         // 625 (exact)
#define TILES_N (NTGT / 16)            // 20  (exact)
#define NUM_TILES (TILES_M * TILES_N)  // 12500
#define WAVES_PER_BLOCK 8
#define THREADS 256

#define KP_DWORDS (NTGT * KP)   // 14400 dwords = 57600 B
#define BX_DWORDS (NTGT * 6)    // 1920 dwords  = 7680 B
#define KP_CHUNKS (KP_DWORDS / 4)  // 3600 x b128
#define BX_CHUNKS (BX_DWORDS / 4)  // 480  x b128

__device__ __forceinline__ void async_wait_zero() {
#if __has_builtin(__builtin_amdgcn_s_wait_asynccnt)
    __builtin_amdgcn_s_wait_asynccnt(0);
#else
    asm volatile("s_wait_asynccnt 0x0" ::: "memory");
#endif
}

// One b128 async DMA: copies 16 bytes from global gbase plus offset goff_byte to LDS byte address lds_byte (GVS mode)
__device__ __forceinline__ void async_g2l_b128(unsigned lds_byte, unsigned goff_byte,
                                               const void* gbase) {
    asm volatile("global_load_async_to_lds_b128 %0, %1, %2"
                 :: "v"(lds_byte), "v"(goff_byte), "s"(gbase)
                 : "memory");
}

__global__ __launch_bounds__(THREADS) void matcher_cost_kernel(
    const float* __restrict__ pred_kp,     // [10000, 45]
    const float* __restrict__ pred_boxes,  // [10000, 6] cxcyczwhd
    const float* __restrict__ pred_conf,   // [10000]
    const float* __restrict__ tgt_boxes,   // [320, 6] cxcyczwhd
    const float* __restrict__ tgt_kp,      // [320, 45]
    float* __restrict__ out)               // [10000, 320]
{
    __shared__ float s_tkp[KP_DWORDS];  // 57600 B
    __shared__ float s_tbx[BX_DWORDS];  // 7680 B

    const int tid = threadIdx.x;

    // ---- Stage all targets into LDS via async global->LDS DMA (ASYNCcnt) ----
    const unsigned tkp_base = (unsigned)(size_t)&s_tkp[0];
    const unsigned tbx_base = (unsigned)(size_t)&s_tbx[0];
    for (int c = tid; c < KP_CHUNKS; c += THREADS)
        async_g2l_b128(tkp_base + (unsigned)c * 16u, (unsigned)c * 16u, tgt_kp);
    for (int c = tid; c < BX_CHUNKS; c += THREADS)
        async_g2l_b128(tbx_base + (unsigned)c * 16u, (unsigned)c * 16u, tgt_boxes);
    async_wait_zero();   // this wave's DMAs landed in LDS
    __syncthreads();     // all waves' DMAs landed

    const int lane = tid & 31;
    const int wave = tid >> 5;
    const int tile = blockIdx.x * WAVES_PER_BLOCK + wave;

    if (tile < NUM_TILES) {  // wave-uniform -> EXEC all-1s inside (WMMA-legal)
        const int tm = tile / TILES_N;
        const int tn = tile - tm * TILES_N;
        const int j    = tn * 16 + (lane & 15);  // this lane's target column
        const int half = lane >> 4;
        const int i0   = tm * 16 + half * 8;     // first pred row for this lane

        // Hint the pred stream into cache (global_prefetch_b8)
        __builtin_prefetch(pred_kp + (size_t)i0 * KP, 0, 1);
        __builtin_prefetch(pred_boxes + (size_t)i0 * 6, 0, 1);

        // ---- Target data from LDS into registers (ds_load) ----
        float tk[KP];
#pragma unroll
        for (int k = 0; k < KP; ++k) tk[k] = s_tkp[j * KP + k];

        float tb[6];
#pragma unroll
        for (int k = 0; k < 6; ++k) tb[k] = s_tbx[j * 6 + k];

        float tlo[3], thi[3];
#pragma unroll
        for (int a = 0; a < 3; ++a) {
            tlo[a] = tb[a] - 0.5f * tb[a + 3];
            thi[a] = tb[a] + 0.5f * tb[a + 3];
        }
        const float vol2 = tb[3] * tb[4] * tb[5];

        // ---- Accumulate 8 rows (WMMA C-layout slice for this lane) ----
        float racc[8];
#pragma unroll
        for (int r = 0; r < 8; ++r) {
            const int gi = i0 + r;
            const float* pk = pred_kp + (size_t)gi * KP;
            const float* pb = pred_boxes + (size_t)gi * 6;

            float s0 = 0.f, s1 = 0.f;
#pragma unroll
            for (int k = 0; k < KP - 1; k += 2) {
                s0 += fabsf(pk[k]     - tk[k]);
                s1 += fabsf(pk[k + 1] - tk[k + 1]);
            }
            s0 += fabsf(pk[KP - 1] - tk[KP - 1]);
            float s = s0 + s1;

            float pbv[6];
#pragma unroll
            for (int k = 0; k < 6; ++k) {
                pbv[k] = pb[k];
                s += fabsf(pbv[k] - tb[k]);
            }

            const float vol1 = pbv[3] * pbv[4] * pbv[5];
            float inter = 1.f, vol_e = 1.f;
#pragma unroll
            for (int a = 0; a < 3; ++a) {
                const float plo = pbv[a] - 0.5f * pbv[a + 3];
                const float phi = pbv[a] + 0.5f * pbv[a + 3];
                inter *= fmaxf(fminf(phi, thi[a]) - fmaxf(plo, tlo[a]), 0.f);
                vol_e *= fmaxf(fmaxf(phi, thi[a]) - fminf(plo, tlo[a]), 0.f);
            }
            const float uni  = vol1 + vol2 - inter;
            const float giou = inter / uni - (vol_e - uni) / vol_e;
            racc[r] = s - giou;  // cost_giou = -giou
        }

        v8f acc;
#pragma unroll
        for (int r = 0; r < 8; ++r) acc[r] = racc[r];

        // ---- Objectness broadcast as rank-1 WMMA: D = A(16x4) * B(4x16) + C ----
        // A[m][k] = -sigmoid(conf_m) for all k (lane<->M map is lane&15, K split
        // across half-waves covered by filling both A VGPRs); B = 0.25 everywhere.
        // D[m][n] = C[m][n] - sigmoid(conf_m), exact in f32.
        const float c = pred_conf[tm * 16 + (lane & 15)];
        const float nsig = -1.f / (1.f + __expf(-c));
        v2f A; A.x = nsig;  A.y = nsig;
        v2f B; B.x = 0.25f; B.y = 0.25f;
        acc = __builtin_amdgcn_wmma_f32_16x16x4_f32(
            false, A, false, B, (short)0, acc, false, false);

        // ---- Store: rows i0..i0+7, fixed column j ----
        float* op = out + (size_t)i0 * NTGT + j;
#pragma unroll
        for (int r = 0; r < 8; ++r) op[(size_t)r * NTGT] = acc[r];
    }
}

extern "C" void kernel_launch(void* const* d_in, const int* in_sizes, int n_in,
                              void* d_out, int out_size, void* d_ws, size_t ws_size,
                              hipStream_t stream) {
    const float* pred_kp    = (const float*)d_in[0];  // (16,625,45)
    const float* pred_boxes = (const float*)d_in[1];  // (16,625,6)
    const float* pred_conf  = (const float*)d_in[2];  // (16,625,1)
    const float* tgt_boxes  = (const float*)d_in[3];  // (320,6)
    const float* tgt_kp     = (const float*)d_in[4];  // (320,45)
    float* out = (float*)d_out;                       // (16,625,320)

    const int blocks = (NUM_TILES + WAVES_PER_BLOCK - 1) / WAVES_PER_BLOCK; // 1563
    matcher_cost_kernel<<<blocks, THREADS, 0, stream>>>(
        pred_kp, pred_boxes, pred_conf, tgt_boxes, tgt_kp, out);
}